// Net_36197984371163
// MI455X (gfx1250) — compile-verified
//
#include <hip/hip_runtime.h>
#include <cstdint>
#include <cstddef>

// SNN LIF scan, MI455X (gfx1250).
// Memory-bound scan: 1 thread == 1 batch element, T=256 sequential steps.
// CDNA5 features used:
//   * global_load_async_to_lds_b64 th:TH_LOAD_NT (ASYNCcnt), 8-deep prefetch
//     pipeline for the x[t] stream (hides HBM latency under the sequential
//     LIF dependence chain; ~2x read-level parallelism vs DEPTH=4)
//   * s_wait_asynccnt / s_wait_dscnt split dependency counters
//   * nontemporal b64 streaming stores for the write-once outputs

#define T_STEPS   256
#define BATCH_N   65536
#define BLOCK     256
#define DEPTH     8                      // async prefetch depth (ring buffer)
#define TILE_F    (BLOCK * 2)            // floats per step per block
#define TILE_B    (TILE_F * 4)           // bytes per step per block (2048)

typedef float v2f __attribute__((ext_vector_type(2)));

__device__ __forceinline__ void issue_async_x_load(uint32_t laddr, uint64_t gaddr) {
  // VDST = per-lane LDS byte offset, VADDR = per-lane 64-bit global address.
  // NT hint: x is read exactly once; don't let it occupy L2.
  asm volatile("global_load_async_to_lds_b64 %0, %1, off th:TH_LOAD_NT"
               :: "v"(laddr), "v"(gaddr) : "memory");
}

__global__ __launch_bounds__(BLOCK) void snn_lif_scan(
    const float* __restrict__ x,
    const float* __restrict__ W1, const float* __restrict__ B1,
    const float* __restrict__ W2, const float* __restrict__ B2,
    float* __restrict__ spk_out, float* __restrict__ mem_out)
{
  __shared__ float tile[DEPTH][TILE_F];   // 16 KiB ring buffer of x tiles

  const int tid = threadIdx.x;
  const int b   = blockIdx.x * BLOCK + tid;          // batch element

  // Flat shared pointer: low 32 bits == workgroup-relative LDS byte offset.
  const uint32_t lds0  = (uint32_t)(uintptr_t)(&tile[0][tid * 2]);
  const uint64_t xbase = (uint64_t)(uintptr_t)x + (uint64_t)b * 8ull;

  // Prologue: fill the DEPTH-deep async pipeline (steps 0..DEPTH-1).
#pragma unroll
  for (int t = 0; t < DEPTH; ++t)
    issue_async_x_load(lds0 + (uint32_t)t * TILE_B,
                       xbase + (uint64_t)t * (BATCH_N * 8ull));

  // Uniform weights -> scalar loads.
  float w1[8], c1[4], w2[8], c2[2];
#pragma unroll
  for (int i = 0; i < 8; ++i) w1[i] = W1[i];
#pragma unroll
  for (int i = 0; i < 4; ++i) c1[i] = B1[i];
#pragma unroll
  for (int i = 0; i < 8; ++i) w2[i] = W2[i];
#pragma unroll
  for (int i = 0; i < 2; ++i) c2[i] = B2[i];

  float m1[4] = {0.f, 0.f, 0.f, 0.f};
  float m2[2] = {0.f, 0.f};

  for (int t = 0; t < T_STEPS; ++t) {
    // Retire the oldest in-flight async load (the one feeding this step).
    // Steady state: DEPTH outstanding -> wait <= DEPTH-1. Tail: drain all.
    if (t + DEPTH <= T_STEPS)
      asm volatile("s_wait_asynccnt 7" ::: "memory");
    else
      asm volatile("s_wait_asynccnt 0" ::: "memory");

    const int slot = t & (DEPTH - 1);
    const float x0 = tile[slot][tid * 2 + 0];
    const float x1 = tile[slot][tid * 2 + 1];

    // DS reads must land in VGPRs before this slot is overwritten by the
    // async engine (DScnt and ASYNCcnt are unordered).
    asm volatile("s_wait_dscnt 0" ::: "memory");
    if (t + DEPTH < T_STEPS)
      issue_async_x_load(lds0 + (uint32_t)slot * TILE_B,
                         xbase + (uint64_t)(t + DEPTH) * (BATCH_N * 8ull));

    // ---- layer 1: fc1 + LIF (4 hidden units) ----
    float s1[4];
#pragma unroll
    for (int h = 0; h < 4; ++h) {
      const float cur = fmaf(w1[2 * h], x0, fmaf(w1[2 * h + 1], x1, c1[h]));
      const float rst = (m1[h] > 1.0f) ? 1.0f : 0.0f;   // reset from PREVIOUS mem
      m1[h] = fmaf(0.95f, m1[h], cur - rst);            // beta*mem + cur - reset*thr
      s1[h] = (m1[h] > 1.0f) ? 1.0f : 0.0f;             // spike on mem_new - 1 > 0
    }

    // ---- layer 2: fc2 + LIF (2 outputs) ----
    v2f sv, mv;
#pragma unroll
    for (int o = 0; o < 2; ++o) {
      float cur = c2[o];
#pragma unroll
      for (int h = 0; h < 4; ++h) cur = fmaf(s1[h], w2[4 * o + h], cur);
      const float rst = (m2[o] > 1.0f) ? 1.0f : 0.0f;
      m2[o] = fmaf(0.95f, m2[o], cur - rst);
      sv[o] = (m2[o] > 1.0f) ? 1.0f : 0.0f;
      mv[o] = m2[o];
    }

    // Streaming (nontemporal) b64 stores; write-once data, keep out of caches.
    const size_t oidx = ((size_t)t * BATCH_N + (size_t)b) * 2;
    __builtin_nontemporal_store(sv, (v2f*)(spk_out + oidx));
    __builtin_nontemporal_store(mv, (v2f*)(mem_out + oidx));
  }
}

extern "C" void kernel_launch(void* const* d_in, const int* in_sizes, int n_in,
                              void* d_out, int out_size, void* d_ws, size_t ws_size,
                              hipStream_t stream) {
  (void)in_sizes; (void)n_in; (void)out_size; (void)d_ws; (void)ws_size;
  const float* x  = (const float*)d_in[0];   // [256, 65536, 2]
  const float* W1 = (const float*)d_in[1];   // [4, 2]
  const float* B1 = (const float*)d_in[2];   // [4]
  const float* W2 = (const float*)d_in[3];   // [2, 4]
  const float* B2 = (const float*)d_in[4];   // [2]

  float* out = (float*)d_out;
  float* spk = out;                                          // spk2_rec flat
  float* mem = out + (size_t)T_STEPS * BATCH_N * 2;          // mem2_rec flat

  dim3 grid(BATCH_N / BLOCK);   // 256 blocks x 256 threads = 65536 threads
  dim3 block(BLOCK);
  snn_lif_scan<<<grid, block, 0, stream>>>(x, W1, B1, W2, B2, spk, mem);
}